// InformationPropagtionLayer_75153337745451
// MI455X (gfx1250) — compile-verified
//
#include <hip/hip_runtime.h>

typedef __attribute__((ext_vector_type(2))) float v2f;
typedef __attribute__((ext_vector_type(8))) float v8f;
typedef int v4i __attribute__((vector_size(16)));   // matches builtin param type

#define C_DIM 16
#define N_DIM 2048
#define D_DIM 64
#define BM    128            // rows of A per workgroup (4 waves x 2 tiles x 16)
#define KC    32             // K chunk per pipeline stage
#define LDA   36             // sA row stride (floats): KC + 4 pad (bank spread)
#define NPAIR (KC / 2)       // 16 K-pair rows in sX
#define LDP   160            // sX pair-row stride: 128 + 32 pad (half-waves hit disjoint bank halves)

#define AS1 __attribute__((address_space(1)))
#define AS3 __attribute__((address_space(3)))

#if defined(__AMDGCN__) && __has_builtin(__builtin_amdgcn_global_load_async_to_lds_b128)
#define USE_ASYNC 1
#endif

__device__ __forceinline__ void async_copy_b128(const float* g, float* l) {
#if defined(USE_ASYNC)
  // gfx1250 async DMA: global -> LDS, 16B per lane, tracked by ASYNCcnt
  __builtin_amdgcn_global_load_async_to_lds_b128((AS1 v4i*)g, (AS3 v4i*)l, 0, 0);
#else
  *(float4*)l = *(const float4*)g;   // synchronous fallback
#endif
}

__device__ __forceinline__ void wait_async_all() {
#if defined(USE_ASYNC)
#if __has_builtin(__builtin_amdgcn_s_wait_asynccnt)
  __builtin_amdgcn_s_wait_asynccnt(0);
#else
  asm volatile("s_wait_asynccnt 0x0" ::: "memory");
#endif
#endif
}

// Stage A chunk (BM x KC) into LDS: 1024 b128, 8 per thread (128 threads).
__device__ __forceinline__ void stage_A(const float* __restrict__ Ac, float* __restrict__ sA,
                                        int rowBase, int kc, int tid) {
  #pragma unroll
  for (int i = 0; i < 8; ++i) {
    int t = tid + i * 128;           // 0..1023
    int r = t >> 3;                  // row 0..127
    int q = (t & 7) << 2;            // float4 slot within the 32-float row chunk
    async_copy_b128(Ac + (size_t)(rowBase + r) * N_DIM + kc + q, sA + r * LDA + q);
  }
}

// Stage X chunk (KC x D) into LDS, K-pair interleaved: element (k,n) -> sX[(k/2)*LDP + 2n + (k&1)].
// Each task loads rows 2p,2p+1 (4 cols) and writes 4 interleaved b64s.
__device__ __forceinline__ void stage_X(const float* __restrict__ Xc, float* __restrict__ sX,
                                        int kc, int tid) {
  #pragma unroll
  for (int i = 0; i < 2; ++i) {
    int t = tid + i * 128;           // 0..255
    int p = t >> 4;                  // K pair 0..15
    int q = (t & 15) << 2;           // col base 0,4,...,60
    float4 x0 = *(const float4*)(Xc + (size_t)(kc + 2 * p) * D_DIM + q);
    float4 x1 = *(const float4*)(Xc + (size_t)(kc + 2 * p + 1) * D_DIM + q);
    float* base = sX + p * LDP + 2 * q;
    ((v2f*)base)[0] = v2f{x0.x, x1.x};
    ((v2f*)base)[1] = v2f{x0.y, x1.y};
    ((v2f*)base)[2] = v2f{x0.z, x1.z};
    ((v2f*)base)[3] = v2f{x0.w, x1.w};
  }
}

#define WMMA_F32(acc, a, b)                                                     \
  acc = __builtin_amdgcn_wmma_f32_16x16x4_f32(false, a, false, b, (short)0,     \
                                              acc, false, false)

// out = 0.5*h_in + 0.5 * dinv[row] * (x + acc), dinv = 1/(1+rowsum(A))
__device__ __forceinline__ void epilogue_tile(const v8f* acc, float rs,
                                              const float* __restrict__ x,
                                              const float* __restrict__ h_in,
                                              float* __restrict__ out,
                                              int c, int m0, int lane) {
  const int mrow = lane & 15;
  float full = rs + __shfl_xor(rs, 16, 32);     // combine half-wave K partitions
  #pragma unroll
  for (int v = 0; v < 8; ++v) {
    const int row = v + ((lane < 16) ? 0 : 8);  // C/D layout: VGPR v -> M=v / v+8
    const float dinv = 1.0f / (1.0f + __shfl(full, row, 32));
    #pragma unroll
    for (int j = 0; j < 4; ++j) {
      size_t idx = ((size_t)c * N_DIM + (m0 + row)) * D_DIM + j * 16 + mrow;
      out[idx] = 0.5f * h_in[idx] + 0.5f * (dinv * (x[idx] + acc[j][v]));
    }
  }
}

__global__ __launch_bounds__(128) void ipl_wmma_kernel(
    const float* __restrict__ x,
    const float* __restrict__ h_in,
    const float* __restrict__ A,
    float* __restrict__ out) {
  __shared__ float sA[2][BM * LDA];      // double-buffered A chunk (2 x 18KB)
  __shared__ float sX[2][NPAIR * LDP];   // double-buffered X chunk, pair-interleaved (2 x 10KB)

  const int c       = blockIdx.x;
  const int rowBase = blockIdx.y * BM;
  const int tid     = threadIdx.x;       // 0..127 (4 wave32 waves)
  const int lane    = tid & 31;
  const int wv      = tid >> 5;          // wave 0..3: owns M-tiles wv and wv+4
  const int mrow    = lane & 15;
  const int koff    = (lane < 16) ? 0 : 2;  // ISA A/B fragment K sub-offset
  const int psel    = (lane < 16) ? 0 : 1;  // which K-pair row of sX this half-wave reads

  const float* Ac = A + (size_t)c * N_DIM * N_DIM;
  const float* Xc = x + (size_t)c * N_DIM * D_DIM;

  v8f acc0[4] = {};                      // M-tile wv
  v8f acc1[4] = {};                      // M-tile wv+4
  float rs0 = 0.0f, rs1 = 0.0f;          // per-lane partial rowsums of A

  // ---- pipeline prologue: stage chunk 0 ----
  stage_A(Ac, &sA[0][0], rowBase, 0, tid);
  stage_X(Xc, &sX[0][0], 0, tid);
  wait_async_all();
  __syncthreads();

  int buf = 0;
  for (int kc = 0; kc < N_DIM; kc += KC) {
    // ---- issue async DMA of next chunk into the other buffer (overlaps WMMA) ----
    if (kc + KC < N_DIM) {
      stage_A(Ac, &sA[buf ^ 1][0], rowBase, kc + KC, tid);
      stage_X(Xc, &sX[buf ^ 1][0], kc + KC, tid);
    }

    // ---- WMMA over current chunk: 8 K-steps x (2 M-tiles x 4 col-tiles) ----
    const float* a0Base = &sA[buf][(wv * 16 + mrow) * LDA + koff];
    const float* a1Base = &sA[buf][((wv + 4) * 16 + mrow) * LDA + koff];
    const float* xb     = &sX[buf][0];
    #pragma unroll
    for (int ks = 0; ks < KC; ks += 4) {
      v2f a0 = *(const v2f*)(a0Base + ks);
      v2f a1 = *(const v2f*)(a1Base + ks);
      rs0 += a0.x + a0.y;                 // rowsums for free
      rs1 += a1.x + a1.y;
      const float* bRow = xb + ((ks >> 1) + psel) * LDP + 2 * mrow;
      #pragma unroll
      for (int j = 0; j < 4; ++j) {
        v2f b = *(const v2f*)(bRow + 32 * j);   // single b64 -> adjacent VGPR pair
        WMMA_F32(acc0[j], a0, b);
        WMMA_F32(acc1[j], a1, b);
      }
    }

    wait_async_all();                     // next buffer's DMA done (overlapped)
    __syncthreads();                      // everyone finished reading current buffer
    buf ^= 1;
  }

  // ---- epilogue ----
  epilogue_tile(acc0, rs0, x, h_in, out, c, rowBase + wv * 16, lane);
  epilogue_tile(acc1, rs1, x, h_in, out, c, rowBase + (wv + 4) * 16, lane);
}

extern "C" void kernel_launch(void* const* d_in, const int* in_sizes, int n_in,
                              void* d_out, int out_size, void* d_ws, size_t ws_size,
                              hipStream_t stream) {
  (void)in_sizes; (void)n_in; (void)d_ws; (void)ws_size; (void)out_size;
  const float* x    = (const float*)d_in[0];
  const float* h_in = (const float*)d_in[1];
  const float* A    = (const float*)d_in[2];
  float* out        = (float*)d_out;

  dim3 grid(C_DIM, N_DIM / BM);   // (16, 16) = 256 workgroups
  dim3 block(128);                // 4 wave32 waves
  ipl_wmma_kernel<<<grid, block, 0, stream>>>(x, h_in, A, out);
}